// LightGCN_50053548868037
// MI455X (gfx1250) — compile-verified
//
#include <hip/hip_runtime.h>
#include <hip/hip_bf16.h>

#define N_USERS 8000
#define N_TOT   16000
#define DD      64
#define TXTD    768
#define NNZE    (1u << 20)
#define NE      (N_TOT * DD)      /* 1,024,000 */

typedef __attribute__((ext_vector_type(2))) float v2f;
typedef __attribute__((ext_vector_type(8))) float v8f;

__device__ __forceinline__ v8f wmma4(v2f a, v2f b, v8f c) {
  return __builtin_amdgcn_wmma_f32_16x16x4_f32(false, a, false, b, (short)0, c,
                                               false, false);
}

// monotone float<->uint encoding for atomic min/max
__device__ __forceinline__ unsigned f_ord(float f) {
  unsigned u = __float_as_uint(f);
  return (u & 0x80000000u) ? ~u : (u | 0x80000000u);
}
__device__ __forceinline__ float ord_f(unsigned o) {
  unsigned u = (o & 0x80000000u) ? (o & 0x7fffffffu) : ~o;
  return __uint_as_float(u);
}

// ---------------------------------------------------------------- elementwise
__global__ void blend_k(const float* __restrict__ user_e,
                        const float* __restrict__ item_e,
                        const float* __restrict__ pos_t, float* __restrict__ e0) {
  int idx = blockIdx.x * blockDim.x + threadIdx.x;
  int row = idx >> 6, d = idx & 63;
  int pr = (row < N_USERS) ? row : row - N_USERS;
  const float* base = (row < N_USERS) ? user_e : item_e;
  e0[idx] = pos_t[pr * DD + d] * 0.6f + 0.4f * base[pr * DD + d];
}

__global__ void zero_k(float* __restrict__ p) {
  p[blockIdx.x * blockDim.x + threadIdx.x] = 0.0f;
}

__global__ void init_mm_k(unsigned* mm) { mm[0] = 0xFFFFFFFFu; mm[1] = 0u; }

// ------------------------------------------------------- SpMM (1 wave / edge)
__global__ void __launch_bounds__(256) spmm_k(const int* __restrict__ rows,
                                              const int* __restrict__ cols,
                                              const float* __restrict__ vals,
                                              const float* __restrict__ x,
                                              float* __restrict__ y) {
  unsigned e = (blockIdx.x * 256u + threadIdx.x) >> 5;
  int lane = threadIdx.x & 31;
  if (e >= NNZE) return;
  int r = rows[e], c = cols[e];
  float v = vals[e];
  unsafeAtomicAdd(&y[r * DD + lane], v * x[c * DD + lane]);
  unsafeAtomicAdd(&y[r * DD + lane + 32], v * x[c * DD + lane + 32]);
}

// ------------------------------------------------- mean of 3 hops + min/max
__global__ void meanminmax_k(const float* __restrict__ a,
                             const float* __restrict__ b,
                             const float* __restrict__ c,
                             float* __restrict__ mean, unsigned* __restrict__ mm) {
  int idx = blockIdx.x * blockDim.x + threadIdx.x;
  float m = (a[idx] + b[idx] + c[idx]) * (1.0f / 3.0f);
  mean[idx] = m;
  float lo = m, hi = m;
#pragma unroll
  for (int k = 1; k <= 16; k <<= 1) {
    lo = fminf(lo, __shfl_xor(lo, k, 32));
    hi = fmaxf(hi, __shfl_xor(hi, k, 32));
  }
  if ((threadIdx.x & 31) == 0) {
    atomicMin(&mm[0], f_ord(lo));
    atomicMax(&mm[1], f_ord(hi));
  }
}

__global__ void normalize_k(const float* __restrict__ mean,
                            const unsigned* __restrict__ mm,
                            float* __restrict__ light) {
  int idx = blockIdx.x * blockDim.x + threadIdx.x;
  float mn = ord_f(mm[0]), mx = ord_f(mm[1]);
  light[idx] = (mean[idx] - mn) / (mx - mn);
}

// ------------------- txt = att @ Wtxt.T + btxt  (K=768)
// 1 wave per 16-row stripe, computes all four 16x16 n-tiles (A loaded once).
// 1000 waves = 125 blocks x 8 waves.
__global__ void __launch_bounds__(256) gemm_txt_k(const float* __restrict__ uatt,
                                                  const float* __restrict__ matt,
                                                  const float* __restrict__ W,
                                                  const float* __restrict__ bias,
                                                  float* __restrict__ out) {
  int mt = blockIdx.x * 8 + (threadIdx.x >> 5);
  int lane = threadIdx.x & 31;
  int half = lane >> 4, lm = lane & 15;
  int m = mt * 16 + lm;
  const float* arow = (m < N_USERS) ? &uatt[(size_t)m * TXTD]
                                    : &matt[(size_t)(m - N_USERS) * TXTD];
  const float* w0 = &W[(lm +  0) * TXTD];
  const float* w1 = &W[(lm + 16) * TXTD];
  const float* w2 = &W[(lm + 32) * TXTD];
  const float* w3 = &W[(lm + 48) * TXTD];
  v8f c0 = {}, c1 = {}, c2 = {}, c3 = {};
  for (int k0 = 0; k0 < TXTD; k0 += 4) {
    int kb = k0 + half * 2;
    v2f a  = *(const v2f*)&arow[kb];
    v2f b0 = *(const v2f*)&w0[kb];
    v2f b1 = *(const v2f*)&w1[kb];
    v2f b2 = *(const v2f*)&w2[kb];
    v2f b3 = *(const v2f*)&w3[kb];
    c0 = wmma4(a, b0, c0);
    c1 = wmma4(a, b1, c1);
    c2 = wmma4(a, b2, c2);
    c3 = wmma4(a, b3, c3);
  }
  float bv0 = bias[lm], bv1 = bias[lm + 16], bv2 = bias[lm + 32], bv3 = bias[lm + 48];
#pragma unroll
  for (int r = 0; r < 8; ++r) {
    int row = mt * 16 + r + half * 8;
    out[row * DD + lm +  0] = c0[r] + bv0;
    out[row * DD + lm + 16] = c1[r] + bv1;
    out[row * DD + lm + 32] = c2[r] + bv2;
    out[row * DD + lm + 48] = c3[r] + bv3;
  }
}

// ------------------- y = x @ W.T + b  (64x64, K=64), x read once per stripe
__global__ void __launch_bounds__(256) gemm64_k(const float* __restrict__ x,
                                                const float* __restrict__ W,
                                                const float* __restrict__ bias,
                                                float* __restrict__ out) {
  int mt = blockIdx.x * 8 + (threadIdx.x >> 5);
  int lane = threadIdx.x & 31;
  int half = lane >> 4, lm = lane & 15;
  const float* arow = &x[(mt * 16 + lm) * DD];
  const float* w0 = &W[(lm +  0) * DD];
  const float* w1 = &W[(lm + 16) * DD];
  const float* w2 = &W[(lm + 32) * DD];
  const float* w3 = &W[(lm + 48) * DD];
  v8f c0 = {}, c1 = {}, c2 = {}, c3 = {};
#pragma unroll
  for (int k0 = 0; k0 < DD; k0 += 4) {
    int kb = k0 + half * 2;
    v2f a  = *(const v2f*)&arow[kb];
    c0 = wmma4(a, *(const v2f*)&w0[kb], c0);
    c1 = wmma4(a, *(const v2f*)&w1[kb], c1);
    c2 = wmma4(a, *(const v2f*)&w2[kb], c2);
    c3 = wmma4(a, *(const v2f*)&w3[kb], c3);
  }
  float bv0 = bias[lm], bv1 = bias[lm + 16], bv2 = bias[lm + 32], bv3 = bias[lm + 48];
#pragma unroll
  for (int r = 0; r < 8; ++r) {
    int row = mt * 16 + r + half * 8;
    out[row * DD + lm +  0] = c0[r] + bv0;
    out[row * DD + lm + 16] = c1[r] + bv1;
    out[row * DD + lm + 32] = c2[r] + bv2;
    out[row * DD + lm + 48] = c3[r] + bv3;
  }
}

// -------------------- flash attention: out = softmax(q k^T) v + light
// 8 waves/block, each wave owns 16 query rows; K/V tiles double-buffered:
// tile j+1 prefetched to registers while tile j is consumed from LDS.
__global__ void __launch_bounds__(256) attn_k(const float* __restrict__ Q,
                                              const float* __restrict__ K,
                                              const float* __restrict__ V,
                                              const float* __restrict__ light,
                                              float* __restrict__ out) {
  __shared__ float kt[16 * 68];       // 16x64 key tile, stride 68 (16B aligned)
  __shared__ float vt[16 * 68];       // 16x64 value tile
  __shared__ float pl[8 * 16 * 18];   // per-wave p staging, stride 18

  const int tid = threadIdx.x;
  const int wv = tid >> 5;
  const int lane = tid & 31;
  const int half = lane >> 4;
  const int lm = lane & 15;
  const int qrow = blockIdx.x * 128 + wv * 16 + lm;

  // preload 16 q A-operands (full K=64 per wave)
  v2f aq[16];
#pragma unroll
  for (int c = 0; c < 16; ++c)
    aq[c] = *(const v2f*)&Q[qrow * DD + c * 4 + half * 2];

  v8f o0 = {}, o1 = {}, o2 = {}, o3 = {};
  float mrun[8], lrun[8];
#pragma unroll
  for (int r = 0; r < 8; ++r) { mrun[r] = -3.0e38f; lrun[r] = 0.0f; }

  const int crow = tid >> 4;          // cooperative loader: row 0..15
  const int ccol = (tid & 15) * 4;    // col 0,4,...,60
  float* plw = &pl[wv * 16 * 18];

  const int NTILE = N_TOT / 16;
  // prefetch tile 0
  float4 k4 = *(const float4*)&K[crow * DD + ccol];
  float4 v4 = *(const float4*)&V[crow * DD + ccol];

  for (int j = 0; j < NTILE; ++j) {
    // commit prefetched tile to LDS
    *(float4*)&kt[crow * 68 + ccol] = k4;
    *(float4*)&vt[crow * 68 + ccol] = v4;
    __syncthreads();

    // prefetch next tile (wait lands at next iteration's LDS commit)
    if (j + 1 < NTILE) {
      k4 = *(const float4*)&K[((j + 1) * 16 + crow) * DD + ccol];
      v4 = *(const float4*)&V[((j + 1) * 16 + crow) * DD + ccol];
    }

    // s = q @ k^T  (B[kb][n] = kt[n][kb])
    v8f s = {};
#pragma unroll
    for (int c = 0; c < 16; ++c) {
      v2f b = *(const v2f*)&kt[lm * 68 + c * 4 + half * 2];
      s = wmma4(aq[c], b, s);
    }

    // per-row max (reduce across 16 lanes of each half)
    float rmax[8];
#pragma unroll
    for (int r = 0; r < 8; ++r) rmax[r] = s[r];
#pragma unroll
    for (int msk = 1; msk <= 8; msk <<= 1)
#pragma unroll
      for (int r = 0; r < 8; ++r)
        rmax[r] = fmaxf(rmax[r], __shfl_xor(rmax[r], msk, 32));

    float scale[8], rsum[8];
#pragma unroll
    for (int r = 0; r < 8; ++r) {
      float mn = fmaxf(mrun[r], rmax[r]);
      scale[r] = __expf(mrun[r] - mn);
      mrun[r] = mn;
      float p = __expf(s[r] - mn);
      s[r] = p;
      rsum[r] = p;
    }
#pragma unroll
    for (int msk = 1; msk <= 8; msk <<= 1)
#pragma unroll
      for (int r = 0; r < 8; ++r)
        rsum[r] += __shfl_xor(rsum[r], msk, 32);
#pragma unroll
    for (int r = 0; r < 8; ++r) {
      lrun[r] = lrun[r] * scale[r] + rsum[r];
      o0[r] *= scale[r]; o1[r] *= scale[r];
      o2[r] *= scale[r]; o3[r] *= scale[r];
    }

    // stage p: C-layout -> row-major LDS -> A-layout
#pragma unroll
    for (int r = 0; r < 8; ++r)
      plw[(r + half * 8) * 18 + lm] = s[r];
    asm volatile("s_wait_dscnt 0" ::: "memory");

    // o += p @ v  (K = 16, 4 chunks)
#pragma unroll
    for (int c = 0; c < 4; ++c) {
      int kb = c * 4 + half * 2;
      v2f a = *(const v2f*)&plw[lm * 18 + kb];
      v2f b0 = {vt[kb * 68 + lm +  0], vt[(kb + 1) * 68 + lm +  0]};
      v2f b1 = {vt[kb * 68 + lm + 16], vt[(kb + 1) * 68 + lm + 16]};
      v2f b2 = {vt[kb * 68 + lm + 32], vt[(kb + 1) * 68 + lm + 32]};
      v2f b3 = {vt[kb * 68 + lm + 48], vt[(kb + 1) * 68 + lm + 48]};
      o0 = wmma4(a, b0, o0);
      o1 = wmma4(a, b1, o1);
      o2 = wmma4(a, b2, o2);
      o3 = wmma4(a, b3, o3);
    }
    __syncthreads();
  }

#pragma unroll
  for (int r = 0; r < 8; ++r) {
    int row = blockIdx.x * 128 + wv * 16 + r + half * 8;
    float il = 1.0f / lrun[r];
    out[row * DD + lm +  0] = o0[r] * il + light[row * DD + lm +  0];
    out[row * DD + lm + 16] = o1[r] * il + light[row * DD + lm + 16];
    out[row * DD + lm + 32] = o2[r] * il + light[row * DD + lm + 32];
    out[row * DD + lm + 48] = o3[r] * il + light[row * DD + lm + 48];
  }
}

// ------------------------------------------------------------------- launch
extern "C" void kernel_launch(void* const* d_in, const int* in_sizes, int n_in,
                              void* d_out, int out_size, void* d_ws, size_t ws_size,
                              hipStream_t stream) {
  const float* user_e = (const float*)d_in[0];
  const float* item_e = (const float*)d_in[1];
  const float* pos_e  = (const float*)d_in[2];
  const float* u_att  = (const float*)d_in[3];
  const float* m_att  = (const float*)d_in[4];
  const float* e_vals = (const float*)d_in[5];
  const float* Wq = (const float*)d_in[6];
  const float* bq = (const float*)d_in[7];
  const float* Wk = (const float*)d_in[8];
  const float* bk = (const float*)d_in[9];
  const float* Wv = (const float*)d_in[10];
  const float* bv = (const float*)d_in[11];
  const float* Wtxt = (const float*)d_in[12];
  const float* btxt = (const float*)d_in[13];
  const int* e_rows = (const int*)d_in[14];
  const int* e_cols = (const int*)d_in[15];
  float* out = (float*)d_out;

  float* ws = (float*)d_ws;
  float* e0    = ws + (size_t)0 * NE;
  float* e1    = ws + (size_t)1 * NE;
  float* e2    = ws + (size_t)2 * NE;
  float* mean  = ws + (size_t)3 * NE;
  float* light = ws + (size_t)4 * NE;
  float* txt   = ws + (size_t)5 * NE;
  float* q     = ws + (size_t)6 * NE;
  float* kk    = ws + (size_t)7 * NE;
  float* vvb   = ws + (size_t)8 * NE;
  unsigned* mm = (unsigned*)(ws + (size_t)9 * NE);

  const int EB = NE / 256;           // 4000 blocks for elementwise
  const int SB = NNZE * 32 / 256;    // 131072 blocks for spmm

  for (int t = 0; t < 2; ++t) {
    blend_k<<<EB, 256, 0, stream>>>(user_e, item_e,
                                    pos_e + (size_t)t * N_USERS * DD, e0);
    zero_k<<<EB, 256, 0, stream>>>(e1);
    spmm_k<<<SB, 256, 0, stream>>>(e_rows + (size_t)t * NNZE,
                                   e_cols + (size_t)t * NNZE,
                                   e_vals + (size_t)t * NNZE, e0, e1);
    zero_k<<<EB, 256, 0, stream>>>(e2);
    spmm_k<<<SB, 256, 0, stream>>>(e_rows + (size_t)t * NNZE,
                                   e_cols + (size_t)t * NNZE,
                                   e_vals + (size_t)t * NNZE, e1, e2);
    init_mm_k<<<1, 1, 0, stream>>>(mm);
    meanminmax_k<<<EB, 256, 0, stream>>>(e0, e1, e2, mean, mm);
    normalize_k<<<EB, 256, 0, stream>>>(mean, mm, light);

    gemm_txt_k<<<125, 256, 0, stream>>>(u_att + (size_t)t * N_USERS * TXTD,
                                        m_att + (size_t)t * N_USERS * TXTD,
                                        Wtxt, btxt, txt);
    gemm64_k<<<125, 256, 0, stream>>>(txt, Wq, bq, q);
    gemm64_k<<<125, 256, 0, stream>>>(light, Wk, bk, kk);
    gemm64_k<<<125, 256, 0, stream>>>(light, Wv, bv, vvb);

    attn_k<<<125, 256, 0, stream>>>(q, kk, vvb, light, out + (size_t)t * NE);
  }
}